// NonLocalMP_72756745994765
// MI455X (gfx1250) — compile-verified
//
#include <hip/hip_runtime.h>

typedef __attribute__((ext_vector_type(2))) float v2f;
typedef __attribute__((ext_vector_type(8))) float v8f;

#define NEG_SLOPE 0.2f

__device__ __forceinline__ float leaky(float x) {
    // x>=0 -> x ; x<0 -> 0.2x  ==  max(x, 0.2x)
    return fmaxf(x, NEG_SLOPE * x);
}

constexpr int B_ = 8;
constexpr int N_ = 16384;
constexpr int V_ = 12;
constexpr int POS = B_ * N_;                    // 131072 positions
constexpr int WPB = 8;                          // waves per block (256 threads)
constexpr int BLOCKS1 = POS / WPB;              // 16384
constexpr int TOTAL = B_ * V_ * N_;             // 1572864 outputs

// ---------------------------------------------------------------------------
// Kernel 1: fused pair-MLP (WMMA f32 16x16x4) + j-sum + 1x1 conv, emits raw y
// and deterministic per-block fp64 partial sums for BatchNorm statistics.
// One wave == one (b,n) position. 144 pairs = 9 WMMA groups of 16 pairs.
// ---------------------------------------------------------------------------
__global__ __launch_bounds__(256) void k_fused(
    const float* __restrict__ F,
    const float* __restrict__ W1, const float* __restrict__ b1,
    const float* __restrict__ W2, const float* __restrict__ b2,
    const float* __restrict__ W3, const float* __restrict__ b3,
    const float* __restrict__ cw, const float* __restrict__ cb,
    float* __restrict__ out, double* __restrict__ ws)
{
    __shared__ float hbuf[WPB][160];        // 144 h3 values + 16 dump slots per wave
    __shared__ float yred[WPB * V_];        // per-block y values for BN partials

    const int tid  = threadIdx.x;
    const int wave = tid >> 5;
    const int lane = tid & 31;
    const int m    = lane & 15;
    const bool hi  = lane >= 16;

    const int q = blockIdx.x * WPB + wave;  // flat (b,n) position
    const int b = q >> 14;                  // N_ == 16384
    const int n = q & (N_ - 1);
    const float* f = F + (size_t)q * V_;

    // --- Build loop-invariant A operands (augmented weights) -------------
    // A(16x4): lanes 0-15 = rows M (VGPR0=K0, VGPR1=K1); lanes 16-31 = (K2=bias, K3=0)
    float w100 = W1[0], w101 = W1[1], w110 = W1[2], w111 = W1[3];
    float w200 = W2[0], w201 = W2[1], w210 = W2[2], w211 = W2[3];
    float w30  = W3[0], w31  = W3[1];
    float bb10 = b1[0], bb11 = b1[1], bb20 = b2[0], bb21 = b2[1], bb30 = b3[0];

    v2f A1, A2, A3;
    A1[0] = hi ? (m == 0 ? bb10 : (m == 1 ? bb11 : 0.f))
               : (m == 0 ? w100 : (m == 1 ? w110 : 0.f));
    A1[1] = hi ? 0.f : (m == 0 ? w101 : (m == 1 ? w111 : 0.f));
    A2[0] = hi ? (m == 0 ? bb20 : (m == 1 ? bb21 : 0.f))
               : (m == 0 ? w200 : (m == 1 ? w210 : 0.f));
    A2[1] = hi ? 0.f : (m == 0 ? w201 : (m == 1 ? w211 : 0.f));
    A3[0] = hi ? (m == 0 ? bb30 : 0.f) : (m == 0 ? w30 : 0.f);
    A3[1] = hi ? 0.f : (m == 0 ? w31 : 0.f);

    const v8f cz = {};

    // --- 9 groups x 3 chained WMMAs ---------------------------------------
    #pragma unroll
    for (int g = 0; g < 9; ++g) {
        const int P = g * 16 + m;           // pair index 0..143
        const int i = P / 12;
        const int j = P - i * 12;
        const float f0 = f[j];              // channel 0 = F[j]
        const float f1 = f[i];              // channel 1 = F[i]

        // B(4x16): lanes0-15 VGPR0=K0 row, VGPR1=K1 row; lanes16-31 = (K2=1, K3=0)
        v2f Bm;
        Bm[0] = hi ? 1.0f : f0;
        Bm[1] = hi ? 0.0f : f1;

        v8f d = __builtin_amdgcn_wmma_f32_16x16x4_f32(
            false, A1, false, Bm, (short)0, cz, false, false);

        v2f B2v;
        B2v[0] = hi ? 1.0f : leaky(d[0]);   // D row0 (lanes0-15) == next K0 row
        B2v[1] = hi ? 0.0f : leaky(d[1]);   // D row1 == next K1 row
        d = __builtin_amdgcn_wmma_f32_16x16x4_f32(
            false, A2, false, B2v, (short)0, cz, false, false);

        v2f B3v;
        B3v[0] = hi ? 1.0f : leaky(d[0]);
        B3v[1] = hi ? 0.0f : leaky(d[1]);
        d = __builtin_amdgcn_wmma_f32_16x16x4_f32(
            false, A3, false, B3v, (short)0, cz, false, false);

        const float h3 = leaky(d[0]);       // scalar MLP output per pair
        const int sidx = hi ? (144 + m) : P; // hi lanes -> dump slots (no divergence)
        hbuf[wave][sidx] = h3;
    }

    __syncthreads();

    // --- j-sum, 1x1 conv, raw y store, BN partials ------------------------
    if (lane < V_) {
        float ms = 0.f;
        #pragma unroll
        for (int j = 0; j < V_; ++j) ms += hbuf[wave][lane * V_ + j];
        const float y = cw[0] * f[lane] + cw[1] * ms + cb[0];
        out[((size_t)b * V_ + lane) * N_ + n] = y;   // [B,V,N] raw (pre-BN)
        yred[wave * V_ + lane] = y;
    }
    __syncthreads();

    if (tid == 0) {                          // fixed-order -> deterministic
        double s = 0.0, s2 = 0.0;
        for (int k = 0; k < WPB * V_; ++k) {
            const double v = (double)yred[k];
            s += v; s2 += v * v;
        }
        ws[2 * blockIdx.x]     = s;
        ws[2 * blockIdx.x + 1] = s2;
    }
}

// ---------------------------------------------------------------------------
// Kernel 2: single block, fixed-order reduction of 16384 fp64 partials.
// ---------------------------------------------------------------------------
__global__ __launch_bounds__(256) void k_stats(double* __restrict__ ws)
{
    __shared__ double ss[256], ss2[256];
    const int t = threadIdx.x;
    double s = 0.0, s2 = 0.0;
    for (int k = t; k < BLOCKS1; k += 256) { s += ws[2 * k]; s2 += ws[2 * k + 1]; }
    ss[t] = s; ss2[t] = s2;
    __syncthreads();
    for (int off = 128; off > 0; off >>= 1) {
        if (t < off) { ss[t] += ss[t + off]; ss2[t] += ss2[t + off]; }
        __syncthreads();
    }
    if (t == 0) {
        const double M    = (double)TOTAL;
        const double mean = ss[0] / M;
        const double var  = ss2[0] / M - mean * mean;   // biased variance
        ws[2 * BLOCKS1]     = mean;
        ws[2 * BLOCKS1 + 1] = 1.0 / sqrt(var + 1e-5);
    }
}

// ---------------------------------------------------------------------------
// Kernel 3: normalize + affine + leaky, in place over d_out.
// ---------------------------------------------------------------------------
__global__ __launch_bounds__(256) void k_bn(
    float* __restrict__ out, const double* __restrict__ ws,
    const float* __restrict__ gamma, const float* __restrict__ beta)
{
    const size_t idx = (size_t)blockIdx.x * 256 + threadIdx.x;
    const float mean = (float)ws[2 * BLOCKS1];
    const float istd = (float)ws[2 * BLOCKS1 + 1];
    const float g = gamma[0], be = beta[0];
    const float x = out[idx];
    out[idx] = leaky((x - mean) * istd * g + be);
}

// ---------------------------------------------------------------------------
extern "C" void kernel_launch(void* const* d_in, const int* in_sizes, int n_in,
                              void* d_out, int out_size, void* d_ws, size_t ws_size,
                              hipStream_t stream)
{
    (void)in_sizes; (void)n_in; (void)out_size; (void)ws_size;
    const float* F     = (const float*)d_in[0];
    const float* W1    = (const float*)d_in[1];
    const float* b1    = (const float*)d_in[2];
    const float* W2    = (const float*)d_in[3];
    const float* b2    = (const float*)d_in[4];
    const float* W3    = (const float*)d_in[5];
    const float* b3    = (const float*)d_in[6];
    const float* cw    = (const float*)d_in[7];
    const float* cb    = (const float*)d_in[8];
    const float* gamma = (const float*)d_in[9];
    const float* beta  = (const float*)d_in[10];
    float*  out = (float*)d_out;
    double* ws  = (double*)d_ws;   // 2*16384 partials + mean + invstd (~256 KB)

    k_fused<<<BLOCKS1, 256, 0, stream>>>(F, W1, b1, W2, b2, W3, b3, cw, cb, out, ws);
    k_stats<<<1, 256, 0, stream>>>(ws);
    k_bn<<<TOTAL / 256, 256, 0, stream>>>(out, ws, gamma, beta);
}